// WordleModel_74423193305129
// MI455X (gfx1250) — compile-verified
//
#include <hip/hip_runtime.h>
#include <math.h>

typedef __attribute__((ext_vector_type(2))) float v2f;
typedef __attribute__((ext_vector_type(8))) float v8f;

#define NWORDS   100000
#define HID      1482
#define HPAD     1488          // HID padded with zeros so WMMA K-tail is exact
#define K2_BLOCKS 782          // ceil(100000 / (8 waves * 16 rows))
#define K4_BLOCKS 391          // ceil(100000 / 256)

// workspace layout (float offsets)
#define WS_H     0             // h1482 + zero pad  [1488]
#define WS_IA    1488          // ia activations    [26]
#define WS_PMAX  1536          // per-block maxes   [782]
#define WS_PSUM  2336          // per-block sums    [391]
#define WS_SCAL  2752          // [0]=gmax [1]=gsum

// ---------------------------------------------------------------------------
// Front: branch linears -> h57 -> h1482 (W1 @ h57 + b1), plus ia to ws.
// Tiny (85K MACs) -> one block.
// ---------------------------------------------------------------------------
__global__ void k_front(const float* __restrict__ x0, const float* __restrict__ x1,
                        const float* __restrict__ x2,
                        const float* __restrict__ Wv, const float* __restrict__ bv,
                        const float* __restrict__ Wi, const float* __restrict__ bi,
                        const float* __restrict__ Wc, const float* __restrict__ bc,
                        const float* __restrict__ W1, const float* __restrict__ b1,
                        float* __restrict__ ws) {
  __shared__ float h57[57];
  const int t = threadIdx.x;
  if (t < 26) {                       // va
    float s = bv[t];
    for (int j = 0; j < 26; ++j) s += Wv[t * 26 + j] * x0[j];
    h57[t] = s;
  } else if (t < 52) {                // ia (also needed for word masking)
    const int r = t - 26;
    float s = bi[r];
    for (int j = 0; j < 26; ++j) s += Wi[r * 26 + j] * x1[j];
    h57[t] = s;
    ws[WS_IA + r] = s;
  } else if (t < 57) {                // cw
    const int r = t - 52;
    float s = bc[r];
    for (int j = 0; j < 5; ++j) s += Wc[r * 5 + j] * x2[j];
    h57[t] = s;
  }
  if (t < HPAD - HID) ws[WS_H + HID + t] = 0.0f;   // zero pad for WMMA K-tail
  __syncthreads();
  for (int r = t; r < HID; r += blockDim.x) {
    float s = b1[r];
    const float* wr = W1 + r * 57;
    for (int j = 0; j < 57; ++j) s += wr[j] * h57[j];
    ws[WS_H + r] = s;
  }
}

// ---------------------------------------------------------------------------
// GEMV logits = W2 @ h + b2 via V_WMMA_F32_16X16X4_F32.
// Wave owns 16 rows. A-tile (16x4 f32, 2 VGPRs): lanes 0-15 -> rows m,
// K = {k0,k0+1}; lanes 16-31 -> rows m, K = {k0+2,k0+3}  => one float2
// nontemporal global load per lane per step. B-tile (4x16): same lane
// pattern over the shared h vector, served from LDS. Column 0 of D holds
// the 16 dot products (lane 0: rows 0-7, lane 16: rows 8-15).
// Memory-bound: 593 MB of W2 streamed once, NT so L2 isn't polluted.
// ---------------------------------------------------------------------------
__global__ void k_logits(const float* __restrict__ W2, const float* __restrict__ b2,
                         const float* __restrict__ ws, float* __restrict__ logits,
                         float* __restrict__ pmax) {
  __shared__ float hlds[HPAD];
  __shared__ float red[256];
  const int t = threadIdx.x;
  for (int i = t; i < HPAD; i += 256) hlds[i] = ws[WS_H + i];
  __syncthreads();

  const int wave = t >> 5;
  const int lane = t & 31;
  const int half = lane >> 4;   // 0: K pair {0,1}, 1: K pair {2,3}
  const int m    = lane & 15;   // row within 16-row tile
  const int base = blockIdx.x * 128 + wave * 16;

  float lmax = -INFINITY;
  if (base < NWORDS) {          // wave-uniform: EXEC all-1s for WMMA
    const float* rowp = W2 + (size_t)(base + m) * HID;
    v8f c = {};
#pragma unroll 4
    for (int k0 = 0; k0 < 1480; k0 += 4) {
      const int kk = k0 + 2 * half;
      double ad = __builtin_nontemporal_load((const double*)(rowp + kk));
      v2f a = __builtin_bit_cast(v2f, ad);
      v2f b = *(const v2f*)(hlds + kk);
      c = __builtin_amdgcn_wmma_f32_16x16x4_f32(false, a, false, b, (short)0, c,
                                                false, false);
    }
    {
      // K tail = {1480,1481}; positions 1482/1483 multiply hlds zeros, and the
      // A load address is clamped inside the row (no OOB, finite garbage * 0).
      double ad = __builtin_nontemporal_load((const double*)(rowp + 1480));
      v2f a = __builtin_bit_cast(v2f, ad);
      v2f b = *(const v2f*)(hlds + 1480 + 2 * half);
      c = __builtin_amdgcn_wmma_f32_16x16x4_f32(false, a, false, b, (short)0, c,
                                                false, false);
    }
    if (m == 0) {               // lanes 0 and 16 hold column 0 of D
      const int r0 = base + 8 * half;
#pragma unroll
      for (int v = 0; v < 8; ++v) {
        float val = c[v] + b2[r0 + v];
        logits[r0 + v] = val;
        lmax = fmaxf(lmax, val);
      }
    }
  }
  red[t] = lmax;
  __syncthreads();
  for (int s = 128; s > 0; s >>= 1) {
    if (t < s) red[t] = fmaxf(red[t], red[t + s]);
    __syncthreads();
  }
  if (t == 0) pmax[blockIdx.x] = red[0];
}

// ---------------------------------------------------------------------------
// Softmax support reductions (traffic here is ~1 MB total: noise).
// ---------------------------------------------------------------------------
__global__ void k_rmax(const float* __restrict__ pin, int n, float* __restrict__ out) {
  __shared__ float red[256];
  const int t = threadIdx.x;
  float mx = -INFINITY;
  for (int i = t; i < n; i += 256) mx = fmaxf(mx, pin[i]);
  red[t] = mx;
  __syncthreads();
  for (int s = 128; s > 0; s >>= 1) {
    if (t < s) red[t] = fmaxf(red[t], red[t + s]);
    __syncthreads();
  }
  if (t == 0) out[0] = red[0];
}

__global__ void k_exp(float* __restrict__ logits, const float* __restrict__ scal,
                      float* __restrict__ psum) {
  __shared__ float red[256];
  const int t = threadIdx.x;
  const int i = blockIdx.x * 256 + t;
  const float gmax = scal[0];
  float e = 0.0f;
  if (i < NWORDS) {
    e = expf(logits[i] - gmax);
    logits[i] = e;
  }
  red[t] = e;
  __syncthreads();
  for (int s = 128; s > 0; s >>= 1) {
    if (t < s) red[t] += red[t + s];
    __syncthreads();
  }
  if (t == 0) psum[blockIdx.x] = red[0];
}

__global__ void k_rsum(const float* __restrict__ pin, int n, float* __restrict__ out) {
  __shared__ float red[256];
  const int t = threadIdx.x;
  float s = 0.0f;
  for (int i = t; i < n; i += 256) s += pin[i];
  red[t] = s;
  __syncthreads();
  for (int st = 128; st > 0; st >>= 1) {
    if (t < st) red[t] += red[t + st];
    __syncthreads();
  }
  if (t == 0) out[0] = red[0];
}

// Normalize + word masking: zero word i if any of its 5 letters has ia[l]==0.
__global__ void k_final(float* __restrict__ probs, const float* __restrict__ scal,
                        const int* __restrict__ words, const float* __restrict__ ia) {
  const int i = blockIdx.x * 256 + threadIdx.x;
  if (i >= NWORDS) return;
  const float inv = 1.0f / scal[1];
  bool masked = false;
#pragma unroll
  for (int j = 0; j < 5; ++j) {
    const int w = words[i * 5 + j];
    if (ia[w] == 0.0f) masked = true;
  }
  probs[i] = masked ? 0.0f : probs[i] * inv;
}

// ---------------------------------------------------------------------------
extern "C" void kernel_launch(void* const* d_in, const int* in_sizes, int n_in,
                              void* d_out, int out_size, void* d_ws, size_t ws_size,
                              hipStream_t stream) {
  const float* x0 = (const float*)d_in[0];
  const float* x1 = (const float*)d_in[1];
  const float* x2 = (const float*)d_in[2];
  // d_in[3] = x3 (unused, as in reference)
  const int*   words = (const int*)d_in[4];
  const float* Wv = (const float*)d_in[5];
  const float* bv = (const float*)d_in[6];
  const float* Wi = (const float*)d_in[7];
  const float* bi = (const float*)d_in[8];
  const float* Wc = (const float*)d_in[9];
  const float* bc = (const float*)d_in[10];
  // d_in[11] = Wt, d_in[12] = bt (computed-but-unused branch)
  const float* W1 = (const float*)d_in[13];
  const float* b1 = (const float*)d_in[14];
  const float* W2 = (const float*)d_in[15];
  const float* b2 = (const float*)d_in[16];

  float* out = (float*)d_out;
  float* ws  = (float*)d_ws;

  k_front<<<1, 256, 0, stream>>>(x0, x1, x2, Wv, bv, Wi, bi, Wc, bc, W1, b1, ws);
  k_logits<<<K2_BLOCKS, 256, 0, stream>>>(W2, b2, ws, out, ws + WS_PMAX);
  k_rmax<<<1, 256, 0, stream>>>(ws + WS_PMAX, K2_BLOCKS, ws + WS_SCAL);
  k_exp<<<K4_BLOCKS, 256, 0, stream>>>(out, ws + WS_SCAL, ws + WS_PSUM);
  k_rsum<<<1, 256, 0, stream>>>(ws + WS_PSUM, K4_BLOCKS, ws + WS_SCAL + 1);
  k_final<<<K4_BLOCKS, 256, 0, stream>>>(out, ws + WS_SCAL, words, ws + WS_IA);
}